// TimeAttender_52029233824181
// MI455X (gfx1250) — compile-verified
//
#include <hip/hip_runtime.h>
#include <math.h>

// Problem constants (fixed by setup_inputs)
#define SEQ 2048
#define BAT 32
#define DIM 512
#define KCONV 147
#define EPS 1e-10f

typedef __attribute__((ext_vector_type(2))) float v2f;
typedef __attribute__((ext_vector_type(8))) float v8f;

// ---------------------------------------------------------------------------
// Kernel 1: attended = state @ W_att^T + b_att via V_WMMA_F32_16X16X4_F32
// GEMM: M=32 (batch), N=512, K=512.  attended[m,n] = sum_k state[m,k]*W[n,k]
// One block of 256 threads = 8 waves; 64 output tiles (2 Mtiles x 32 Ntiles),
// 8 tiles per wave, 128 K-steps of 4 per tile.
// A 16x4 f32 layout (ISA 7.12.2): lane&15 = M row; lanes 0-15 hold K={0,1},
// lanes 16-31 hold K={2,3} in the VGPR pair. B assumed symmetric (lane&15 = N).
// C/D: VGPR v -> M = v (lanes 0-15) / v+8 (lanes 16-31), N = lane&15.
// ---------------------------------------------------------------------------
__global__ __launch_bounds__(256)
void k1_attended_wmma(const float* __restrict__ state,
                      const float* __restrict__ W,
                      const float* __restrict__ bias,
                      float* __restrict__ att) {
    const int tid  = threadIdx.x;
    const int wave = tid >> 5;
    const int lane = tid & 31;
    const int half = lane >> 4;   // 0: K 0/1, 1: K 2/3
    const int l    = lane & 15;

    for (int t = 0; t < 8; ++t) {
        const int idx    = wave * 8 + t;
        const int m_base = (idx & 1) * 16;
        const int n_base = (idx >> 1) * 16;

        v8f c = {};
        const float* ap = state + (m_base + l) * DIM + 2 * half;
        const float* bp = W     + (n_base + l) * DIM + 2 * half;

#pragma unroll 4
        for (int k0 = 0; k0 < DIM; k0 += 4) {
            v2f a = *(const v2f*)(ap + k0);
            v2f b = *(const v2f*)(bp + k0);
            c = __builtin_amdgcn_wmma_f32_16x16x4_f32(
                    /*neg_a=*/false, a, /*neg_b=*/false, b,
                    /*c_mod=*/(short)0, c, /*reuse_a=*/false, /*reuse_b=*/false);
        }

        const float bv = bias[n_base + l];
#pragma unroll
        for (int v = 0; v < 8; ++v) {
            const int m = m_base + v + 8 * half;
            att[m * DIM + n_base + l] = c[v] + bv;
        }
    }
}

// ---------------------------------------------------------------------------
// Kernel 2: ns[b,:] = attended[b,:] / safe_norm(attended[b,:])
// One block per b, 256 threads, each owns 2 elements of the 512-row.
// ---------------------------------------------------------------------------
__global__ __launch_bounds__(256)
void k2_normalize(const float* __restrict__ att, float* __restrict__ ns) {
    __shared__ float red[256];
    const int b = blockIdx.x;
    const int t = threadIdx.x;
    const float v0 = att[b * DIM + t];
    const float v1 = att[b * DIM + 256 + t];
    red[t] = v0 * v0 + v1 * v1;
    __syncthreads();
    for (int o = 128; o > 0; o >>= 1) {
        if (t < o) red[t] += red[t + o];
        __syncthreads();
    }
    const float n    = sqrtf(red[0]);
    const float inv  = 1.0f / ((n != 0.0f) ? n : EPS);
    ns[b * DIM + t]       = v0 * inv;
    ns[b * DIM + 256 + t] = v1 * inv;
}

// ---------------------------------------------------------------------------
// Kernel 3: weights[b,s] = dot(ns[b], enc[s,b,:]) / safe_norm(enc[s,b,:])
// One wave32 per (s,b) pair; pair id p = s*32+b so consecutive waves stream
// consecutive 2KB rows. Each lane: 4x float4 contiguous loads (b128), then
// shfl_xor wave reduction. 134 MB streamed once (RT hints -> parks in L2).
// ---------------------------------------------------------------------------
__global__ __launch_bounds__(256)
void k3_weights(const float* __restrict__ enc,
                const float* __restrict__ ns,
                float* __restrict__ wout) {
    const int p    = blockIdx.x * 8 + (threadIdx.x >> 5);  // (s*BAT + b)
    const int lane = threadIdx.x & 31;
    const int b    = p & (BAT - 1);
    const int s    = p >> 5;

    const float4* e4 = (const float4*)(enc + (size_t)p * DIM);
    const float4* n4 = (const float4*)(ns + b * DIM);

    float dot = 0.0f, ssq = 0.0f;
#pragma unroll
    for (int j = 0; j < 4; ++j) {
        const float4 ev = e4[j * 32 + lane];
        const float4 nv = n4[j * 32 + lane];
        dot += ev.x * nv.x + ev.y * nv.y + ev.z * nv.z + ev.w * nv.w;
        ssq += ev.x * ev.x + ev.y * ev.y + ev.z * ev.z + ev.w * ev.w;
    }
#pragma unroll
    for (int o = 16; o > 0; o >>= 1) {
        dot += __shfl_xor(dot, o, 32);
        ssq += __shfl_xor(ssq, o, 32);
    }
    if (lane == 0) {
        const float n = sqrtf(ssq);
        wout[b * SEQ + s] = dot / ((n != 0.0f) ? n : EPS);
    }
}

// ---------------------------------------------------------------------------
// Kernel 4: per-row conv(147) + relu + renorm, softmax(weights), product,
// renormalize -> norm_w. One block per b. Writes norm_w to ws (for k5) and
// to d_out (second output).
// ---------------------------------------------------------------------------
__device__ __forceinline__ float blkSum(float v, float* red) {
    const int t = threadIdx.x;
    __syncthreads();
    red[t] = v;
    __syncthreads();
    for (int o = 128; o > 0; o >>= 1) {
        if (t < o) red[t] += red[t + o];
        __syncthreads();
    }
    const float r = red[0];
    __syncthreads();
    return r;
}
__device__ __forceinline__ float blkMax(float v, float* red) {
    const int t = threadIdx.x;
    __syncthreads();
    red[t] = v;
    __syncthreads();
    for (int o = 128; o > 0; o >>= 1) {
        if (t < o) red[t] = fmaxf(red[t], red[t + o]);
        __syncthreads();
    }
    const float r = red[0];
    __syncthreads();
    return r;
}

__global__ __launch_bounds__(256)
void k4_rowprobs(const float* __restrict__ prev,
                 const float* __restrict__ cw,
                 const float* __restrict__ cb,
                 float* __restrict__ wbuf,      // in: weights, out: norm_w
                 float* __restrict__ out_nw) {  // d_out + BAT*DIM
    __shared__ float sPrev[SEQ];
    __shared__ float sW[KCONV + 13];
    __shared__ float red[256];
    const int b = blockIdx.x;
    const int t = threadIdx.x;

    for (int s = t; s < SEQ; s += 256) sPrev[s] = prev[b * SEQ + s];
    for (int j = t; j < KCONV; j += 256) sW[j] = cw[j];
    __syncthreads();
    const float cbv = cb[0];
    const int HALO = KCONV / 2;  // 73

    float adj[SEQ / 256];
    float local = 0.0f;
#pragma unroll
    for (int i = 0; i < SEQ / 256; ++i) {
        const int s  = i * 256 + t;
        const int lo = (s - HALO < 0) ? 0 : s - HALO;
        const int hi = (s + HALO > SEQ - 1) ? SEQ - 1 : s + HALO;
        float sum = 0.0f;
        for (int x = lo; x <= hi; ++x) sum += sPrev[x] * sW[x - s + HALO];
        float a = sum + cbv;
        a = (a > 0.0f) ? a : 0.0f;
        adj[i] = a;
        local += a;
    }
    const float sum_adj = blkSum(local, red);

    float wl[SEQ / 256];
    float mx = -3.402823466e38f;
#pragma unroll
    for (int i = 0; i < SEQ / 256; ++i) {
        wl[i] = wbuf[b * SEQ + i * 256 + t];
        mx = fmaxf(mx, wl[i]);
    }
    const float wmax = blkMax(mx, red);

    float esum = 0.0f;
#pragma unroll
    for (int i = 0; i < SEQ / 256; ++i) {
        wl[i] = __expf(wl[i] - wmax);
        esum += wl[i];
    }
    const float sum_e = blkSum(esum, red);

    float tsum = 0.0f;
#pragma unroll
    for (int i = 0; i < SEQ / 256; ++i) {
        const float tv = (adj[i] / sum_adj) * (wl[i] / sum_e);
        adj[i] = tv;
        tsum += tv;
    }
    const float sum_t = blkSum(tsum, red);
    const float invt  = 1.0f / sum_t;

#pragma unroll
    for (int i = 0; i < SEQ / 256; ++i) {
        const int s   = i * 256 + t;
        const float v = adj[i] * invt;
        wbuf[b * SEQ + s]   = v;
        out_nw[b * SEQ + s] = v;
    }
}

// ---------------------------------------------------------------------------
// Kernel 5: partial applied[b,d] over an s-chunk. Block = (b, dchunk, schunk),
// 256 threads own 256 consecutive d -> each s-iteration reads 1KB contiguous.
// Second pass over enc: should hit the 192MB L2 (RT residency from pass 1).
// Deterministic (no atomics): partials reduced in k6.
// ---------------------------------------------------------------------------
__global__ __launch_bounds__(256)
void k5_apply_partial(const float* __restrict__ enc,
                      const float* __restrict__ nw,
                      float* __restrict__ part, int NS) {
    const int id = blockIdx.x;
    const int sc = id % NS;
    const int dc = (id / NS) & 1;           // DIM/256 = 2 chunks
    const int b  = id / (NS * 2);
    const int d  = dc * 256 + threadIdx.x;
    const int sLen = SEQ / NS;
    const int s0   = sc * sLen;

    float acc = 0.0f;
#pragma unroll 4
    for (int s = s0; s < s0 + sLen; ++s) {
        acc += nw[b * SEQ + s] * enc[((size_t)s * BAT + b) * DIM + d];
    }
    part[(((size_t)b * 2 + dc) * NS + sc) * 256 + threadIdx.x] = acc;
}

__global__ __launch_bounds__(256)
void k6_reduce(const float* __restrict__ part, float* __restrict__ out, int NS) {
    const int i  = blockIdx.x * 256 + threadIdx.x;  // 0 .. BAT*DIM
    const int b  = i >> 9;
    const int d  = i & (DIM - 1);
    const int dc = d >> 8;
    const int t  = d & 255;
    float sum = 0.0f;
    for (int sc = 0; sc < NS; ++sc)
        sum += part[(((size_t)b * 2 + dc) * NS + sc) * 256 + t];
    out[i] = sum;
}

// ---------------------------------------------------------------------------
extern "C" void kernel_launch(void* const* d_in, const int* in_sizes, int n_in,
                              void* d_out, int out_size, void* d_ws, size_t ws_size,
                              hipStream_t stream) {
    const float* enc   = (const float*)d_in[0];  // [S,B,D]
    const float* state = (const float*)d_in[1];  // [B,D]
    const float* prev  = (const float*)d_in[2];  // [B,S]
    const float* Watt  = (const float*)d_in[3];  // [D,D]
    const float* batt  = (const float*)d_in[4];  // [D]
    const float* convw = (const float*)d_in[5];  // [1,1,K]
    const float* convb = (const float*)d_in[6];  // [1]

    float* out     = (float*)d_out;          // applied [B,D] then norm_w [B,S]
    float* out_nw  = out + BAT * DIM;

    // Workspace layout (floats)
    float* ws      = (float*)d_ws;
    float* ws_att  = ws;                          // BAT*DIM
    float* ws_ns   = ws_att + BAT * DIM;          // BAT*DIM
    float* ws_w    = ws_ns + BAT * DIM;           // BAT*SEQ (weights -> norm_w)
    float* ws_part = ws_w + BAT * SEQ;            // BAT*DIM*NS

    int NS = 8;
    {
        const size_t base  = (size_t)(2 * BAT * DIM + BAT * SEQ);
        const size_t need8 = (base + (size_t)BAT * DIM * 8) * sizeof(float);
        if (ws_size < need8) NS = 1;
    }

    // 1) attended = state @ W^T + b  (WMMA f32 16x16x4)
    k1_attended_wmma<<<dim3(1), dim3(256), 0, stream>>>(state, Watt, batt, ws_att);
    // 2) ns = attended / safe_norm
    k2_normalize<<<dim3(BAT), dim3(256), 0, stream>>>(ws_att, ws_ns);
    // 3) weights[b,s] (streaming pass 1 over enc; wave per (s,b))
    k3_weights<<<dim3(SEQ * BAT / 8), dim3(256), 0, stream>>>(enc, ws_ns, ws_w);
    // 4) conv + softmax + renorm -> norm_w (also second output)
    k4_rowprobs<<<dim3(BAT), dim3(256), 0, stream>>>(prev, convw, convb, ws_w, out_nw);
    // 5) applied partials (streaming pass 2 over enc, L2-resident)
    k5_apply_partial<<<dim3(BAT * 2 * NS), dim3(256), 0, stream>>>(enc, ws_w, ws_part, NS);
    // 6) deterministic reduce of s-chunk partials -> applied
    k6_reduce<<<dim3(BAT * DIM / 256), dim3(256), 0, stream>>>(ws_part, out, NS);
}